// Attention_80702435491904
// MI455X (gfx1250) — compile-verified
//
#include <hip/hip_runtime.h>
#include <hip/hip_bf16.h>
#include <cstddef>

typedef __attribute__((ext_vector_type(2))) float v2f;
typedef __attribute__((ext_vector_type(4))) float v4f;
typedef __attribute__((ext_vector_type(8))) float v8f;

#define DM 2048
#define SEQ 4096
#define NHEADS 16
#define HD 128
#define ROWS 8192            // b*n = 2*4096

#define BM 128
#define BN 128               // == one head
#define BK 16
#define LDSS 20              // padded row stride in floats (80B: 16B aligned, conflict-free)
#define TILEF (BM * LDSS)    // floats per LDS tile (A or B)

#define LN10000_OVER_64 0.14391156831212787f  // ln(10000)/64

#if __has_builtin(__builtin_amdgcn_s_wait_asynccnt)
#define WAIT_ASYNC(n) __builtin_amdgcn_s_wait_asynccnt(n)
#else
#define WAIT_ASYNC(n) asm volatile("s_wait_asynccnt %0" ::"i"(n) : "memory")
#endif

// DMA one 16-byte chunk from global memory straight into LDS (ASYNCcnt-tracked).
__device__ __forceinline__ void async_b128(const float* g, const float* l) {
  unsigned loff =
      (unsigned)(unsigned long long)(__attribute__((address_space(3))) const float*)l;
  asm volatile("global_load_async_to_lds_b128 %0, %1, off"
               :: "v"(loff), "v"(g)
               : "memory");
}

// ---------------------------------------------------------------------------
// LayerNorm: xn = (x - mu) * rsqrt(var + eps) * gamma + beta
// One block per row (2048 floats), 256 threads, fully coalesced b128 traffic.
// ---------------------------------------------------------------------------
__global__ __launch_bounds__(256) void ln_kernel(const float* __restrict__ x,
                                                 const float* __restrict__ gamma,
                                                 const float* __restrict__ beta,
                                                 float* __restrict__ xn) {
  const int row = blockIdx.x;
  const int tid = threadIdx.x;
  const float* xr = x + (size_t)row * DM;
  float* outr = xn + (size_t)row * DM;

  v4f v0 = *(const v4f*)(xr + tid * 4);
  v4f v1 = *(const v4f*)(xr + tid * 4 + 1024);

  float s = 0.f, s2 = 0.f;
#pragma unroll
  for (int i = 0; i < 4; ++i) { s += v0[i] + v1[i]; s2 += v0[i]*v0[i] + v1[i]*v1[i]; }

  // wave32 tree reduce
#pragma unroll
  for (int off = 16; off > 0; off >>= 1) {
    s  += __shfl_down(s,  off, 32);
    s2 += __shfl_down(s2, off, 32);
  }
  __shared__ float red[2][8];
  const int wave = tid >> 5, lane = tid & 31;
  if (lane == 0) { red[0][wave] = s; red[1][wave] = s2; }
  __syncthreads();
  float ts = 0.f, ts2 = 0.f;
#pragma unroll
  for (int i = 0; i < 8; ++i) { ts += red[0][i]; ts2 += red[1][i]; }
  const float mu   = ts * (1.f / DM);
  const float var  = ts2 * (1.f / DM) - mu * mu;
  const float rstd = rsqrtf(var + 1e-5f);

  v4f g0 = *(const v4f*)(gamma + tid * 4);
  v4f g1 = *(const v4f*)(gamma + tid * 4 + 1024);
  v4f b0 = *(const v4f*)(beta + tid * 4);
  v4f b1 = *(const v4f*)(beta + tid * 4 + 1024);
  v4f o0, o1;
#pragma unroll
  for (int i = 0; i < 4; ++i) {
    o0[i] = (v0[i] - mu) * rstd * g0[i] + b0[i];
    o1[i] = (v1[i] - mu) * rstd * g1[i] + b1[i];
  }
  *(v4f*)(outr + tid * 4)        = o0;
  *(v4f*)(outr + tid * 4 + 1024) = o1;
}

// ---------------------------------------------------------------------------
// GEMM + fused RoPE:  out[b,h,n,:] = rope( xn[b,n,:] @ w[e,:]^T ) for one head.
// Workgroup tile 128x128 (8 waves stacked in M), K tiled by 16 through LDS with
// async global->LDS DMA double buffering (ASYNCcnt). V_WMMA_F32_16X16X4_F32.
// K-loop unrolled by 2 so buffer parity is compile-time (no runtime selects in
// the hot loop). RoPE pairs (j, j+64) are accumulator tiles nt and nt+4 in
// identical lanes, so the rotation is register-local.
// ---------------------------------------------------------------------------

// Issue one K-tile's async loads: 2 b128 for A + 2 b128 for B per thread
// => 4 ASYNCcnt-tracked instructions per wave per tile.
__device__ __forceinline__ void issue_tile(const float* __restrict__ xn,
                                           const float* __restrict__ w,
                                           int m0, int e0, int k0,
                                           float* buf, int tid) {
  float* bufA = buf;
  float* bufB = buf + TILEF;
#pragma unroll
  for (int i = 0; i < 2; ++i) {
    const int c   = tid + 256 * i;     // chunk 0..511
    const int r   = c >> 2;            // 0..127
    const int col = (c & 3) * 4;       // 0,4,8,12 (floats)
    async_b128(xn + (size_t)(m0 + r) * DM + k0 + col, bufA + r * LDSS + col);
    async_b128(w  + (size_t)(e0 + r) * DM + k0 + col, bufB + r * LDSS + col);
  }
}

__device__ __forceinline__ void compute_tile(const float* buf, v8f (&acc)[8],
                                             int mrow, int nlan, int koff) {
  const float* ldsA = buf;
  const float* ldsB = buf + TILEF;
#pragma unroll
  for (int ks = 0; ks < BK; ks += 4) {
    const v2f afrag = *(const v2f*)(ldsA + mrow * LDSS + ks + koff);
#pragma unroll
    for (int nt = 0; nt < 8; ++nt) {
      const v2f bfrag = *(const v2f*)(ldsB + (nt * 16 + nlan) * LDSS + ks + koff);
      acc[nt] = __builtin_amdgcn_wmma_f32_16x16x4_f32(
          /*neg_a=*/false, afrag, /*neg_b=*/false, bfrag,
          /*c_mod=*/(short)0, acc[nt], /*reuse_a=*/false, /*reuse_b=*/false);
    }
  }
}

__global__ __launch_bounds__(256) void gemm_rope_kernel(const float* __restrict__ xn,
                                                        const float* __restrict__ w,
                                                        float* __restrict__ out,
                                                        int do_rope) {
  __shared__ __align__(16) float lds[2 * 2 * TILEF];   // 2 buffers x (A+B) = 40 KB

  const int tid  = threadIdx.x;
  const int wave = tid >> 5;          // 0..7 -> 16-row M strip
  const int lane = tid & 31;
  const int m0   = blockIdx.x * BM;   // global row base (b*n flattened)
  const int head = blockIdx.y;        // 0..15
  const int e0   = head * BN;

  v8f acc[8];
  {
    v8f z = {};
#pragma unroll
    for (int t = 0; t < 8; ++t) acc[t] = z;
  }

  const int mrow = wave * 16 + (lane & 15);
  const int nlan = lane & 15;
  const int koff = 2 * (lane >> 4);   // lanes 16-31 handle K+2,K+3

  float* buf0 = lds;                  // [A | B]
  float* buf1 = lds + 2 * TILEF;      // [A | B]

  issue_tile(xn, w, m0, e0, 0, buf0, tid);

  for (int k0 = 0; k0 < DM; k0 += 2 * BK) {
    // ---- phase 0: compute buf0, prefetch k0+BK into buf1 (always valid) ----
    issue_tile(xn, w, m0, e0, k0 + BK, buf1, tid);
    WAIT_ASYNC(4);                    // <=4 outstanding => buf0 landed
    __syncthreads();
    compute_tile(buf0, acc, mrow, nlan, koff);
    __syncthreads();                  // buf0 reads done before it is re-filled

    // ---- phase 1: compute buf1, prefetch k0+2*BK into buf0 ----
    if (k0 + 2 * BK < DM) {
      issue_tile(xn, w, m0, e0, k0 + 2 * BK, buf0, tid);
      WAIT_ASYNC(4);                  // <=4 outstanding => buf1 landed
    } else {
      WAIT_ASYNC(0);
    }
    __syncthreads();
    compute_tile(buf1, acc, mrow, nlan, koff);
    __syncthreads();
  }

  // Epilogue: C/D layout -> lane 0-15: M=r, N=lane; lane 16-31: M=r+8, N=lane-16
  const int mbase = m0 + wave * 16 + 8 * (lane >> 4);

  if (do_rope) {
#pragma unroll
    for (int nt = 0; nt < 4; ++nt) {
      const int j = nt * 16 + nlan;                       // 0..63
      const float invf = __expf(-(float)j * LN10000_OVER_64);
#pragma unroll
      for (int r = 0; r < 8; ++r) {
        const int pos = (mbase + r) & (SEQ - 1);          // n within batch
        float sn, cs;
        __sincosf((float)pos * invf, &sn, &cs);
        const float a0 = acc[nt][r];
        const float a1 = acc[nt + 4][r];
        acc[nt][r]     = a0 * cs - a1 * sn;
        acc[nt + 4][r] = a1 * cs + a0 * sn;
      }
    }
  }

  // Store in (b, h, n, hd) layout.
#pragma unroll
  for (int nt = 0; nt < 8; ++nt) {
    const int j = nt * 16 + nlan;
#pragma unroll
    for (int r = 0; r < 8; ++r) {
      const int m   = mbase + r;
      const int b   = m >> 12;                            // /4096
      const int pos = m & (SEQ - 1);
      const size_t o = (((size_t)(b * NHEADS + head)) * SEQ + pos) * HD + j;
      out[o] = acc[nt][r];
    }
  }
}

// ---------------------------------------------------------------------------
extern "C" void kernel_launch(void* const* d_in, const int* in_sizes, int n_in,
                              void* d_out, int out_size, void* d_ws, size_t ws_size,
                              hipStream_t stream) {
  const float* x     = (const float*)d_in[0];
  const float* gamma = (const float*)d_in[1];
  const float* beta  = (const float*)d_in[2];
  const float* wq    = (const float*)d_in[3];
  const float* wk    = (const float*)d_in[4];
  const float* wv    = (const float*)d_in[5];
  // d_in[6] = num_heads (fixed 16 for this shape)

  float* out = (float*)d_out;
  float* xnb = (float*)d_ws;            // 8192*2048*4 = 64 MB, stages in L2

  ln_kernel<<<ROWS, 256, 0, stream>>>(x, gamma, beta, xnb);

  const size_t per_mat = (size_t)2 * NHEADS * SEQ * HD; // 16,777,216 elems
  dim3 grid(ROWS / BM, NHEADS);
  gemm_rope_kernel<<<grid, 256, 0, stream>>>(xnb, wq, out,               1);
  gemm_rope_kernel<<<grid, 256, 0, stream>>>(xnb, wk, out + per_mat,     1);
  gemm_rope_kernel<<<grid, 256, 0, stream>>>(xnb, wv, out + 2 * per_mat, 0);
}